// EnhancedGNN_26491358282017
// MI455X (gfx1250) — compile-verified
//
#include <hip/hip_runtime.h>
#include <hip/hip_bf16.h>
#include <math.h>

typedef __attribute__((ext_vector_type(2))) float v2f;
typedef __attribute__((ext_vector_type(8))) float v8f;

#define DH 128   // hidden/feature dim (d_in == d_h == 128 in the reference)

// ---------------------------------------------------------------------------
// init: deg = 1.0 (self-loop contribution), zero both scatter accumulators
// ---------------------------------------------------------------------------
__global__ void gcn_init(float* __restrict__ deg, float* __restrict__ agg1,
                         float* __restrict__ agg2, int n_nodes) {
    int idx = blockIdx.x * blockDim.x + threadIdx.x;
    int total = n_nodes * DH;
    if (idx < total) { agg1[idx] = 0.0f; agg2[idx] = 0.0f; }
    if (idx < n_nodes) deg[idx] = 1.0f;
}

// deg[dst] += 1 per edge (self-loops already baked in as the 1.0 init)
__global__ void gcn_degree(const long long* __restrict__ dst,
                           float* __restrict__ deg, int E) {
    int e = blockIdx.x * blockDim.x + threadIdx.x;
    if (e < E) atomicAdd(&deg[(int)dst[e]], 1.0f);
}

// dinv = rsqrt(deg); deg >= 1 always (self-loop), so no zero guard needed
__global__ void gcn_rsqrt(float* __restrict__ deg, int n) {
    int i = blockIdx.x * blockDim.x + threadIdx.x;
    if (i < n) deg[i] = rsqrtf(deg[i]);
}

// ---------------------------------------------------------------------------
// C[rows x 128] = A[rows x 128] @ W[128 x 128], fp32 WMMA 16x16x4.
// One 16x16 output tile per wave32; block = 256 threads = 8 waves covering the
// 8 column tiles of one 16-row stripe (blockIdx.x = row tile). rows % 16 == 0
// so EXEC is all-1s everywhere (WMMA requirement).
//
// VGPR layouts (ISA 7.12.2, 32-bit):
//   A 16x4 : lane half 0 -> K={k0,k0+1}, half 1 -> K={k0+2,k0+3}; M = lane&15
//   B 4x16 : mirrored, N = lane&15
//   C/D    : VGPR r holds row (r + 8*half), col = lane&15
// ---------------------------------------------------------------------------
__global__ void gcn_gemm(const float* __restrict__ A, const float* __restrict__ W,
                         float* __restrict__ C) {
    const int wave = threadIdx.x >> 5;
    const int lane = threadIdx.x & 31;
    const int half = lane >> 4;
    const int lid  = lane & 15;
    const int m0 = blockIdx.x << 4;
    const int n0 = wave << 4;
    const int row = m0 + lid;
    const int col = n0 + lid;

    const float* __restrict__ arow = A + (size_t)row * DH;

    v8f acc = {};
    #pragma unroll 8
    for (int k0 = 0; k0 < DH; k0 += 4) {
        // A fragment: two consecutive K values for this lane's half
        v2f a = *(const v2f*)(arow + k0 + 2 * half);
        // B fragment: same two K rows of W at this lane's column
        v2f b;
        b.x = W[(size_t)(k0 + 2 * half + 0) * DH + col];
        b.y = W[(size_t)(k0 + 2 * half + 1) * DH + col];
        acc = __builtin_amdgcn_wmma_f32_16x16x4_f32(
            /*neg_a=*/false, a, /*neg_b=*/false, b,
            /*c_mod=*/(short)0, acc, /*reuse_a=*/false, /*reuse_b=*/false);
    }

    #pragma unroll
    for (int r = 0; r < 8; ++r)
        C[(size_t)(m0 + r + 8 * half) * DH + col] = acc[r];
}

// ---------------------------------------------------------------------------
// One wave per edge: agg[dst] += h[src] * (dinv[src]*dinv[dst])
// 128 feats / 32 lanes = float4 per lane; atomics resolve in L2 (agg fits).
// ---------------------------------------------------------------------------
__global__ void gcn_scatter(const float* __restrict__ h,
                            const long long* __restrict__ src,
                            const long long* __restrict__ dst,
                            const float* __restrict__ dinv,
                            float* __restrict__ agg, int E) {
    int e = blockIdx.x * (blockDim.x >> 5) + (threadIdx.x >> 5);
    if (e >= E) return;
    int lane = threadIdx.x & 31;
    int s = (int)src[e];
    int d = (int)dst[e];
    float nrm = dinv[s] * dinv[d];
    float4 v = ((const float4*)(h + (size_t)s * DH))[lane];
    float* op = agg + (size_t)d * DH + lane * 4;
    atomicAdd(op + 0, v.x * nrm);
    atomicAdd(op + 1, v.y * nrm);
    atomicAdd(op + 2, v.z * nrm);
    atomicAdd(op + 3, v.w * nrm);
}

// agg = relu(agg + h * dinv^2 (self-loop) + bias), in place -> layer-2 input
__global__ void gcn_epilogue_relu(float* __restrict__ agg, const float* __restrict__ h,
                                  const float* __restrict__ dinv,
                                  const float* __restrict__ bias, int n_nodes) {
    int idx = blockIdx.x * blockDim.x + threadIdx.x;
    if (idx >= n_nodes * DH) return;
    int i = idx >> 7;           // / 128
    int c = idx & (DH - 1);     // % 128
    float di = dinv[i];
    float v = agg[idx] + h[idx] * di * di + bias[c];
    agg[idx] = fmaxf(v, 0.0f);
}

// ---------------------------------------------------------------------------
// Final: per node, v2 = agg2 + h2*dinv^2 + b2; out = sigmoid(dot(v2, Wl) + bl)
// One wave per node, float4 per lane, wave32 shuffle reduction.
// ---------------------------------------------------------------------------
__global__ void gcn_final(const float* __restrict__ agg2, const float* __restrict__ h2,
                          const float* __restrict__ dinv, const float* __restrict__ b2,
                          const float* __restrict__ Wl, const float* __restrict__ bl,
                          float* __restrict__ out, int n_nodes) {
    int i = blockIdx.x * (blockDim.x >> 5) + (threadIdx.x >> 5);
    if (i >= n_nodes) return;
    int lane = threadIdx.x & 31;
    float di = dinv[i];
    float sc = di * di;
    float4 a  = ((const float4*)(agg2 + (size_t)i * DH))[lane];
    float4 hh = ((const float4*)(h2 + (size_t)i * DH))[lane];
    float4 bb = ((const float4*)b2)[lane];
    float4 wl = ((const float4*)Wl)[lane];
    float v = (a.x + hh.x * sc + bb.x) * wl.x
            + (a.y + hh.y * sc + bb.y) * wl.y
            + (a.z + hh.z * sc + bb.z) * wl.z
            + (a.w + hh.w * sc + bb.w) * wl.w;
    #pragma unroll
    for (int off = 16; off > 0; off >>= 1)
        v += __shfl_xor(v, off, 32);
    if (lane == 0)
        out[i] = 1.0f / (1.0f + __expf(-(v + bl[0])));
}

// ---------------------------------------------------------------------------
extern "C" void kernel_launch(void* const* d_in, const int* in_sizes, int n_in,
                              void* d_out, int out_size, void* d_ws, size_t ws_size,
                              hipStream_t stream) {
    const float*     x    = (const float*)d_in[0];
    const long long* edge = (const long long*)d_in[1];   // int64 [2, E]
    const float*     W1   = (const float*)d_in[2];
    const float*     b1   = (const float*)d_in[3];
    const float*     W2   = (const float*)d_in[4];
    const float*     b2   = (const float*)d_in[5];
    const float*     Wl   = (const float*)d_in[6];
    const float*     bl   = (const float*)d_in[7];
    float* out = (float*)d_out;

    const int N = in_sizes[0] / DH;   // 50000 (multiple of 16)
    const int E = in_sizes[1] / 2;    // 800000
    const long long* src = edge;
    const long long* dst = edge + E;

    // workspace layout (floats): dinv[N] | h[N*128] | agg1[N*128] | agg2[N*128]
    float* f0 = (float*)d_ws;                 // dinv
    float* f1 = f0 + N;                       // GEMM output (h), reused layer 2
    float* f2 = f1 + (size_t)N * DH;          // agg1 -> layer-2 input (in place)
    float* f3 = f2 + (size_t)N * DH;          // agg2

    const int totNH = N * DH;
    dim3 blk(256);

    // normalization
    gcn_init  <<<(totNH + 255) / 256, blk, 0, stream>>>(f0, f2, f3, N);
    gcn_degree<<<(E + 255) / 256,     blk, 0, stream>>>(dst, f0, E);
    gcn_rsqrt <<<(N + 255) / 256,     blk, 0, stream>>>(f0, N);

    // layer 1: h = x @ W1 ; agg1 = scatter ; relu(agg1 + self + b1)
    gcn_gemm   <<<N / 16,        blk, 0, stream>>>(x, W1, f1);
    gcn_scatter<<<(E + 7) / 8,   blk, 0, stream>>>(f1, src, dst, f0, f2, E);
    gcn_epilogue_relu<<<(totNH + 255) / 256, blk, 0, stream>>>(f2, f1, f0, b1, N);

    // layer 2: h2 = relu_out @ W2 ; agg2 = scatter ; fused epilogue+lin+sigmoid
    gcn_gemm   <<<N / 16,        blk, 0, stream>>>(f2, W2, f1);
    gcn_scatter<<<(E + 7) / 8,   blk, 0, stream>>>(f1, src, dst, f0, f3, E);
    gcn_final  <<<(N + 7) / 8,   blk, 0, stream>>>(f3, f1, f0, b2, Wl, bl, out, N);
}